// ATT_learner_62723702391210
// MI455X (gfx1250) — compile-verified
//
#include <hip/hip_runtime.h>
#include <hip/hip_bf16.h>

#define N 8192
#define ISIZE 256
#define KEEP 31          // K+1 entries kept per row
#define EPS 1e-12f

typedef __attribute__((ext_vector_type(2))) float v2f;
typedef __attribute__((ext_vector_type(8))) float v8f;

// ---------------------------------------------------------------------------
// Kernel 1: e = row_l2_normalize( relu(x * w0) * w1 )
// one block per row, 256 threads = 1 column each
// ---------------------------------------------------------------------------
__global__ __launch_bounds__(256) void prep_normalize(
    const float* __restrict__ feat, const float* __restrict__ w,
    float* __restrict__ e) {
  const int row = blockIdx.x;
  const int col = threadIdx.x;

  float v = feat[row * ISIZE + col] * w[col];   // layer 0 diag scale
  v = fmaxf(v, 0.0f);                           // relu (not after last layer)
  v = v * w[ISIZE + col];                       // layer 1 diag scale

  // sum of squares across the row (wave32 shuffle + LDS combine)
  float ss = v * v;
  #pragma unroll
  for (int o = 16; o > 0; o >>= 1) ss += __shfl_xor(ss, o, 32);

  __shared__ float swave[8];
  const int wave = threadIdx.x >> 5;
  const int lane = threadIdx.x & 31;
  if (lane == 0) swave[wave] = ss;
  __syncthreads();

  float tot = 0.0f;
  #pragma unroll
  for (int i = 0; i < 8; ++i) tot += swave[i];

  const float denom = fmaxf(sqrtf(tot), EPS);
  e[row * ISIZE + col] = v / denom;
}

// ---------------------------------------------------------------------------
// Kernel 2: sim = e @ e^T via V_WMMA_F32_16X16X4_F32
// block = 256 threads (8 waves). Each wave: one 16-row M tile, 4 N tiles (64
// cols). Grid (64,128): x covers M in 128-row superblocks, y covers N in
// 64-col superblocks.
//
// fp32 WMMA lane mapping (per ISA 16x4 A / 4x16 B / 16x16 C):
//   A: lane l holds A[l&15][2*(l>>4) + {0,1}]  (2 VGPRs)
//   B: lane l holds B[2*(l>>4) + {0,1}][l&15]  (2 VGPRs)  ; B = e^T, so
//      b = e[nBase + (l&15)][k0 + 2*(l>>4) + {0,1}]  -- same pattern as A.
//   C: element c[i] -> row = i + 8*(l>>4), col = l&15
// ---------------------------------------------------------------------------
__global__ __launch_bounds__(256) void sim_gemm_wmma(
    const float* __restrict__ e, float* __restrict__ out) {
  const int wave = threadIdx.x >> 5;
  const int lane = threadIdx.x & 31;

  const int mBase = (blockIdx.x * 8 + wave) * 16;  // 64 * 8 * 16 = 8192
  const int nBase = blockIdx.y * 64;               // 128 * 64    = 8192

  const int r  = lane & 15;
  const int kb = (lane >> 4) * 2;

  const float* __restrict__ aPtr = e + (mBase + r) * ISIZE + kb;
  const float* __restrict__ bPtr = e + (nBase + r) * ISIZE + kb;

  v8f c0 = {}, c1 = {}, c2 = {}, c3 = {};

  #pragma unroll 4
  for (int k0 = 0; k0 < ISIZE; k0 += 4) {
    const v2f a  = *(const v2f*)(aPtr + k0);
    const v2f b0 = *(const v2f*)(bPtr + k0);
    const v2f b1 = *(const v2f*)(bPtr + 16 * ISIZE + k0);
    const v2f b2 = *(const v2f*)(bPtr + 32 * ISIZE + k0);
    const v2f b3 = *(const v2f*)(bPtr + 48 * ISIZE + k0);
    c0 = __builtin_amdgcn_wmma_f32_16x16x4_f32(false, a, false, b0, (short)0, c0, false, false);
    c1 = __builtin_amdgcn_wmma_f32_16x16x4_f32(false, a, false, b1, (short)0, c1, false, false);
    c2 = __builtin_amdgcn_wmma_f32_16x16x4_f32(false, a, false, b2, (short)0, c2, false, false);
    c3 = __builtin_amdgcn_wmma_f32_16x16x4_f32(false, a, false, b3, (short)0, c3, false, false);
  }

  const int colBase = nBase + (lane & 15);
  const int rowOff  = (lane >> 4) * 8;
  #pragma unroll
  for (int i = 0; i < 8; ++i) {
    const size_t row = (size_t)(mBase + rowOff + i);
    float* rp = out + row * (size_t)N + colBase;
    rp[0]  = c0[i];
    rp[16] = c1[i];
    rp[32] = c2[i];
    rp[48] = c3[i];
  }
}

// ---------------------------------------------------------------------------
// Kernel 3: per-row top-(K+1) threshold + mask + relu, in place on sim.
// One block (256 threads) per row; row cached in 32 regs/thread.
// 31 rounds of block-wide max extraction -> threshold = 31st largest.
// ---------------------------------------------------------------------------
__global__ __launch_bounds__(256) void topk_relu(float* __restrict__ sim) {
  const int row = blockIdx.x;
  float* __restrict__ rp = sim + (size_t)row * (size_t)N;

  float vals[32];
  #pragma unroll
  for (int i = 0; i < 32; ++i) vals[i] = rp[i * 256 + threadIdx.x];

  __shared__ float smax[8];
  __shared__ int   sclaim;

  const int wave = threadIdx.x >> 5;
  const int lane = threadIdx.x & 31;
  const float NEG = -__builtin_inff();

  float thresh = NEG;
  for (int it = 0; it < KEEP; ++it) {
    // local max over this thread's 32 elements
    float lm = NEG;
    #pragma unroll
    for (int i = 0; i < 32; ++i) lm = fmaxf(lm, vals[i]);

    // wave reduce, then 8-wave LDS combine
    float wm = lm;
    #pragma unroll
    for (int o = 16; o > 0; o >>= 1) wm = fmaxf(wm, __shfl_xor(wm, o, 32));
    if (lane == 0) smax[wave] = wm;
    if (threadIdx.x == 0) sclaim = 0;
    __syncthreads();

    float bm = smax[0];
    #pragma unroll
    for (int i = 1; i < 8; ++i) bm = fmaxf(bm, smax[i]);

    // exactly one thread removes exactly one instance of the max
    if (lm == bm) {
      if (atomicAdd(&sclaim, 1) == 0) {
        bool done = false;
        #pragma unroll
        for (int i = 0; i < 32; ++i) {
          if (!done && vals[i] == bm) { vals[i] = NEG; done = true; }
        }
      }
    }
    thresh = bm;
    __syncthreads();  // protect smax/sclaim reuse next round
  }

  // keep entries >= 31st-largest, then relu (mask-then-relu == this)
  #pragma unroll
  for (int i = 0; i < 32; ++i) {
    const float orig = rp[i * 256 + threadIdx.x];
    rp[i * 256 + threadIdx.x] = (orig >= thresh) ? fmaxf(orig, 0.0f) : 0.0f;
  }
}

// ---------------------------------------------------------------------------
extern "C" void kernel_launch(void* const* d_in, const int* in_sizes, int n_in,
                              void* d_out, int out_size, void* d_ws, size_t ws_size,
                              hipStream_t stream) {
  const float* feat = (const float*)d_in[0];   // [8192, 256] f32
  const float* w    = (const float*)d_in[1];   // [2, 256] f32
  // d_in[2] = edge_ori, unused by the reference output
  float* out = (float*)d_out;                  // [8192, 8192] f32
  float* e   = (float*)d_ws;                   // [8192, 256] f32 scratch (8 MB)

  prep_normalize<<<N, 256, 0, stream>>>(feat, w, e);
  sim_gemm_wmma<<<dim3(64, 128), 256, 0, stream>>>(e, out);
  topk_relu<<<N, 256, 0, stream>>>(out);
}